// SimpleAffinity_7739531067454
// MI455X (gfx1250) — compile-verified
//
#include <hip/hip_runtime.h>

// SimpleAffinity for MI455X (gfx1250, wave32).
//
// out[b,i,j] = exp(-50*max(||f_i-f_j||^2,0)) * exp(-(dy^2+dx^2)/50) * [|dy|<=8 & |dx|<=8]
// where f = 16x bilinear (half-pixel) downsample of images -> [B, 48*48, 3].
//
// Roofline: output 16*2304*2304*4B = 340MB -> ~15us floor @ 23.3TB/s; compute is
// trivial (0.5 GFLOP GEMM + <=30M exps after pruning) -> pure streaming-store kernel.
// One V_WMMA_F32_16X16X4_F32 per 16x16 tile: sq_i folded into A's 4th column,
// sq_j folded into the C init, so D == color distance directly.

typedef float v2f __attribute__((ext_vector_type(2)));
typedef float v8f __attribute__((ext_vector_type(8)));

#define GW    48           // feature grid width/height
#define NN    2304         // GW*GW
#define BB    16           // batch
#define HW    768          // image H = W
#define TILES 144          // NN / 16

// ---------------------------------------------------------------------------
// Kernel 1: build feature tables.
//   Aform[row] = (-2f0, -2f1, -2f2, s)   s = |f|^2   (WMMA A matrix rows)
//   Bform[row] = ( f0,   f1,   f2,  1)               (WMMA B matrix cols)
// 16x bilinear downsample, half-pixel centers -> sample at 16k+7.5 -> mean of
// the 2x2 block at rows/cols {16k+7, 16k+8}.
// ---------------------------------------------------------------------------
__global__ __launch_bounds__(256) void prep_feats(const float* __restrict__ img,
                                                  float* __restrict__ Af,
                                                  float* __restrict__ Bf) {
    int t = blockIdx.x * blockDim.x + threadIdx.x;   // 0 .. BB*NN-1
    if (t >= BB * NN) return;
    int b = t / NN;
    int n = t - b * NN;
    int y = n / GW;
    int x = n - y * GW;
    int py = 16 * y + 7, px = 16 * x + 7;
    const float* base = img + (size_t)b * 3 * HW * HW;
    const float* p0 = base + ((size_t)0 * HW + py) * HW + px;
    const float* p1 = base + ((size_t)1 * HW + py) * HW + px;
    const float* p2 = base + ((size_t)2 * HW + py) * HW + px;
    float f0 = 0.25f * (p0[0] + p0[1] + p0[HW] + p0[HW + 1]);
    float f1 = 0.25f * (p1[0] + p1[1] + p1[HW] + p1[HW + 1]);
    float f2 = 0.25f * (p2[0] + p2[1] + p2[HW] + p2[HW + 1]);
    float s = f0 * f0 + f1 * f1 + f2 * f2;
    *(float4*)(Af + (size_t)t * 4) = make_float4(-2.f * f0, -2.f * f1, -2.f * f2, s);
    *(float4*)(Bf + (size_t)t * 4) = make_float4(f0, f1, f2, 1.0f);
}

// ---------------------------------------------------------------------------
// Kernel 2: one wave32 per 16x16 output tile.
//   A (16x4 f32): lane L -> row (i0 + L&15), comps 2*(L>>4)..+1   (ISA layout)
//   B (4x16 f32): lane L -> col (j0 + L&15), comps 2*(L>>4)..+1
//   C init: s_j broadcast (N is fixed per lane in the 16x16 f32 C layout)
//   D[m][n] = s_i + s_j - 2 f_i.f_j
// 16 | 48 -> each tile has a single y for rows and for cols: dy is tile-uniform,
// so |dy|>8 tiles (65%) take a uniform branch that streams zeros as 2x b128
// stores per wave, keeping EXEC all-ones around the WMMA on the live path.
// ---------------------------------------------------------------------------
__global__ __launch_bounds__(256) void affinity_tiles(const float* __restrict__ Af,
                                                      const float* __restrict__ Bf,
                                                      float* __restrict__ out) {
    const int lane = threadIdx.x & 31;
    const int wave = threadIdx.x >> 5;
    const int tj = blockIdx.x * 8 + wave;   // 0..143
    const int ti = blockIdx.y;              // 0..143
    const int b  = blockIdx.z;              // 0..15
    const int i0 = ti * 16, j0 = tj * 16;
    const int nl = lane & 15, hi = lane >> 4;

    float* otile = out + ((size_t)b * NN + i0) * NN + j0;

    const int yi  = i0 / GW;                // tile-uniform
    const int yj  = j0 / GW;
    const int dy  = yi - yj;
    const int xi0 = i0 - yi * GW;
    const int xj0 = j0 - yj * GW;

    if (dy > 8 || dy < -8) {
        // Whole tile outside spatial support: write 1KB of zeros as 2x b128/lane.
        const int zr = lane >> 2;           // 0..7
        const int zc = (lane & 3) * 4;      // 0,4,8,12  (16B aligned cols)
        const float4 z = make_float4(0.f, 0.f, 0.f, 0.f);
        float* pz = otile + (size_t)zr * NN + zc;
        *(float4*)(pz) = z;
        *(float4*)(pz + (size_t)8 * NN) = z;
        return;
    }

    const float* Ab = Af + (size_t)b * NN * 4;
    const float* Bb = Bf + (size_t)b * NN * 4;

    v2f a   = *(const v2f*)(Ab + (size_t)(i0 + nl) * 4 + 2 * hi);
    v2f bm  = *(const v2f*)(Bb + (size_t)(j0 + nl) * 4 + 2 * hi);
    float sqj = Ab[(size_t)(j0 + nl) * 4 + 3];

    v8f c;
#pragma unroll
    for (int r = 0; r < 8; ++r) c[r] = sqj;

    // D = A(16x4) x B(4x16) + C  -> color distance tile
    c = __builtin_amdgcn_wmma_f32_16x16x4_f32(false, a, false, bm, (short)0, c,
                                              false, false);

    const float dy2  = (float)(dy * dy);
    const int   dx0  = xi0 + 8 * hi - (xj0 + nl);  // dx at r=0, per-lane constant
    float* po = otile + (size_t)(8 * hi) * NN + nl; // one base addr, imm offsets below

#pragma unroll
    for (int r = 0; r < 8; ++r) {
        int   dx   = dx0 + r;
        float dxf  = (float)dx;
        float dist = fmaxf(c[r], 0.0f);
        float arg  = -50.0f * dist - 0.02f * fmaf(dxf, dxf, dy2);
        float e    = __expf(arg);                       // unconditional -> no branch
        bool  inb  = (unsigned)(dx + 8) <= 16u;
        po[r * NN] = inb ? e : 0.0f;                    // v_cndmask, imm-offset store
    }
}

// ---------------------------------------------------------------------------
extern "C" void kernel_launch(void* const* d_in, const int* in_sizes, int n_in,
                              void* d_out, int out_size, void* d_ws, size_t ws_size,
                              hipStream_t stream) {
    (void)in_sizes; (void)n_in; (void)out_size; (void)ws_size;
    const float* img = (const float*)d_in[0];
    float* out = (float*)d_out;

    float* Af = (float*)d_ws;                       // BB*NN*4 floats = 589,824 B
    float* Bf = Af + (size_t)BB * NN * 4;           // same size; total ~1.18 MB

    int total = BB * NN;
    prep_feats<<<(total + 255) / 256, 256, 0, stream>>>(img, Af, Bf);

    dim3 grid(TILES / 8, TILES, BB);                // 8 waves/block, 1 tile/wave
    affinity_tiles<<<grid, 256, 0, stream>>>(Af, Bf, out);
}